// OptLayer_19078244729550
// MI455X (gfx1250) — compile-verified
//
#include <hip/hip_runtime.h>
#include <cstdint>

// ---------------------------------------------------------------------------
// Per-row exact QP projection:  z = clip(nu - x, 0, 35),  sum(z) = 150, D=5.
// Since 4*35 = 140 < 150, no coordinate can sit at the lower bound, so only
// the number m of upper-saturated coordinates (the m smallest x) is unknown.
// Sort 5 values, test the 5 candidates nu_m = (150 - 35m + suffix_sum)/(5-m)
// with division-free validity checks, one f64 division at the end.
//
// Data movement (the part MI455X rewards at 23.3 TB/s):
//   global --async b128--> LDS --(conflict-free stride-5 dword reads)--> f64
//   f64 --> LDS --async b128--> global   (fully coalesced streams)
// ---------------------------------------------------------------------------

#define THREADS 128
#define ROWS_PER_BLOCK 128
#define DWORDS_PER_BLOCK (ROWS_PER_BLOCK * 5)    // 640 dwords
#define BYTES_PER_BLOCK  (DWORDS_PER_BLOCK * 4)  // 2560 bytes = 160 x b128

__device__ __forceinline__ void async_ld_b32(uint32_t lds_byte_off, const void* gaddr) {
  asm volatile("global_load_async_to_lds_b32 %0, %1, off"
               :: "v"(lds_byte_off), "v"((uint64_t)(uintptr_t)gaddr)
               : "memory");
}
__device__ __forceinline__ void async_ld_b128(uint32_t lds_byte_off, const void* gaddr) {
  asm volatile("global_load_async_to_lds_b128 %0, %1, off"
               :: "v"(lds_byte_off), "v"((uint64_t)(uintptr_t)gaddr)
               : "memory");
}
__device__ __forceinline__ void async_st_b32(const void* gaddr, uint32_t lds_byte_off) {
  asm volatile("global_store_async_from_lds_b32 %0, %1, off"
               :: "v"((uint64_t)(uintptr_t)gaddr), "v"(lds_byte_off)
               : "memory");
}
__device__ __forceinline__ void async_st_b128(const void* gaddr, uint32_t lds_byte_off) {
  asm volatile("global_store_async_from_lds_b128 %0, %1, off"
               :: "v"((uint64_t)(uintptr_t)gaddr), "v"(lds_byte_off)
               : "memory");
}
__device__ __forceinline__ void wait_asynccnt0() {
  asm volatile("s_wait_asynccnt 0x0" ::: "memory");
}

__global__ __launch_bounds__(THREADS) void qp_proj_kernel(const float* __restrict__ x,
                                                          float* __restrict__ out,
                                                          int nrows) {
  __shared__ __align__(16) float tin[DWORDS_PER_BLOCK];
  __shared__ __align__(16) float tout[DWORDS_PER_BLOCK];

  const int tid = threadIdx.x;
  const long long row0   = (long long)blockIdx.x * ROWS_PER_BLOCK;
  const long long dword0 = row0 * 5;
  const long long totdw  = (long long)nrows * 5;
  const long long rem    = totdw - dword0;
  const int cnt  = (rem > (long long)DWORDS_PER_BLOCK) ? DWORDS_PER_BLOCK : (int)rem;
  const bool full = (cnt == DWORDS_PER_BLOCK);   // uniform: one branch, no EXEC churn

  const uint32_t lds_in  = (uint32_t)(uintptr_t)tin;   // low 32 bits = LDS offset
  const uint32_t lds_out = (uint32_t)(uintptr_t)tout;
  const char* gin  = (const char*)(x + dword0);
  char*       gout = (char*)(out + dword0);

  // ---- async global -> LDS ----
  if (full) {
    // 2560 B tile as 160 b128 transfers: all 128 threads + first 32 again
    async_ld_b128(lds_in + 16u * (uint32_t)tid, gin + 16 * tid);
    if (tid < 32)
      async_ld_b128(lds_in + 2048u + 16u * (uint32_t)tid, gin + 2048 + 16 * tid);
  } else {
#pragma unroll
    for (int k = 0; k < 5; ++k) {
      const int d = tid + k * THREADS;
      if (d < cnt) async_ld_b32(lds_in + 4u * (uint32_t)d, gin + 4 * d);
    }
  }
  wait_asynccnt0();      // this wave's async loads done (cross-wave deps exist)
  __syncthreads();       // all waves' tiles visible

  const long long row = row0 + tid;
  if (row < nrows) {
    const int base = tid * 5;  // stride-5 dwords: gcd(5,64)=1 -> no bank conflicts
    const double v0 = (double)tin[base + 0];
    const double v1 = (double)tin[base + 1];
    const double v2 = (double)tin[base + 2];
    const double v3 = (double)tin[base + 3];
    const double v4 = (double)tin[base + 4];

    // 5-element sorting network (9 compare-exchanges), ascending a0..a4
    double a0 = v0, a1 = v1, a2 = v2, a3 = v3, a4 = v4;
#define CE(p, q) { double lo_ = fmin(p, q); double hi_ = fmax(p, q); p = lo_; q = hi_; }
    CE(a0, a1) CE(a3, a4) CE(a2, a4) CE(a2, a3) CE(a0, a3)
    CE(a0, a2) CE(a1, a4) CE(a1, a3) CE(a1, a2)
#undef CE

    // suffix sums of sorted values
    const double s4 = a4;
    const double s3 = a3 + s4;
    const double s2 = a2 + s3;
    const double s1 = a1 + s2;
    const double s0 = a0 + s1;

    // candidate numerators: nu_m = (150 - 35m + s_m) / (5 - m)
    const double n0 = 150.0 + s0;
    const double n1 = 115.0 + s1;
    const double n2 =  80.0 + s2;
    const double n3 =  45.0 + s3;
    const double n4 =  10.0 + s4;

    const double b0 = a0 + 35.0, b1 = a1 + 35.0, b2 = a2 + 35.0, b3 = a3 + 35.0;

    // validity (division-free):  (5-m)*b_{m-1} <= num_m <= (5-m)*b_m
    const bool ok0 = (n0 <= 5.0 * b0);
    const bool ok1 = (n1 >= 4.0 * b0) && (n1 <= 4.0 * b1);
    const bool ok2 = (n2 >= 3.0 * b1) && (n2 <= 3.0 * b2);
    const bool ok3 = (n3 >= 2.0 * b2) && (n3 <= 2.0 * b3);
    // m=4: nu = 10 + a4 -> interior z = 10 <= 35 always; only lower check matters

    double num = n0, den = 5.0;
    if (!ok0) {
      if (ok1)      { num = n1; den = 4.0; }
      else if (ok2) { num = n2; den = 3.0; }
      else if (ok3) { num = n3; den = 2.0; }
      else          { num = n4; den = 1.0; }
    }
    const double nu = num / den;

    tout[base + 0] = (float)fmin(fmax(nu - v0, 0.0), 35.0);
    tout[base + 1] = (float)fmin(fmax(nu - v1, 0.0), 35.0);
    tout[base + 2] = (float)fmin(fmax(nu - v2, 0.0), 35.0);
    tout[base + 3] = (float)fmin(fmax(nu - v3, 0.0), 35.0);
    tout[base + 4] = (float)fmin(fmax(nu - v4, 0.0), 35.0);
  }

  __syncthreads();       // all ds_stores to tout complete before async engine reads

  // ---- async LDS -> global ----
  if (full) {
    async_st_b128(gout + 16 * tid, lds_out + 16u * (uint32_t)tid);
    if (tid < 32)
      async_st_b128(gout + 2048 + 16 * tid, lds_out + 2048u + 16u * (uint32_t)tid);
  } else {
#pragma unroll
    for (int k = 0; k < 5; ++k) {
      const int d = tid + k * THREADS;
      if (d < cnt) async_st_b32(gout + 4 * d, lds_out + 4u * (uint32_t)d);
    }
  }
  // S_ENDPGM performs an implicit wait-idle, draining ASYNCcnt for the stores.
}

extern "C" void kernel_launch(void* const* d_in, const int* in_sizes, int n_in,
                              void* d_out, int out_size, void* d_ws, size_t ws_size,
                              hipStream_t stream) {
  (void)n_in; (void)out_size; (void)d_ws; (void)ws_size;
  const float* x = (const float*)d_in[0];
  float* out = (float*)d_out;
  const int nrows = in_sizes[0] / 5;            // 2,000,000
  const int blocks = (nrows + ROWS_PER_BLOCK - 1) / ROWS_PER_BLOCK;  // 15625
  qp_proj_kernel<<<blocks, THREADS, 0, stream>>>(x, out, nrows);
}